// Seq2seqAttention_10187662426575
// MI455X (gfx1250) — compile-verified
//
#include <hip/hip_runtime.h>
#include <math.h>

// Seq2seq attention, fused single-pass (online softmax), MI455X / gfx1250.
// B=32, S=4096, D=2048, all f32. HBM-bound: 1 GiB of encoder states read once.

typedef float v2f __attribute__((ext_vector_type(2)));
typedef float v4f __attribute__((ext_vector_type(4)));
typedef float v8f __attribute__((ext_vector_type(8)));

#define B_ 32
#define S_ 4096
#define D_ 2048
#define TILE_R 16
#define ROW_PAD 4                    // pad rows by 4 words -> conflict-free WMMA-A reads
#define ROW_W (D_ + ROW_PAD)         // 2052 words per LDS row
#define TILE_W (TILE_R * ROW_W)      // 32832 words per tile buffer (131,328 B)
#define NTHREADS 256

// Flat LDS pointer -> 32-bit LDS offset (low 32 bits of the flat address).
__device__ __forceinline__ unsigned lds_off(const void* p) {
  return (unsigned)(unsigned long long)p;
}

// DMA one 16x2048 f32 tile into LDS via CDNA5 async loads (ASYNCcnt-tracked).
// 256 threads x 32 float4 each = 128 KB; per-lane LDS dest applies the row pad.
__device__ __forceinline__ void async_tile_load(const float* gsrc_row0,
                                                unsigned lds_base_bytes, int tid) {
  #pragma unroll
  for (int i = 0; i < 32; ++i) {
    int f = i * NTHREADS + tid;          // float4 index within tile
    int r = f >> 7;                      // 128 float4 per row
    int c4 = f & 127;
    unsigned long long g =
        (unsigned long long)(const void*)(gsrc_row0 + (size_t)r * D_ + (size_t)c4 * 4);
    unsigned l = lds_base_bytes + (unsigned)(r * ROW_W + c4 * 4) * 4u;
    asm volatile("global_load_async_to_lds_b128 %0, %1, off"
                 :: "v"(l), "v"(g) : "memory");
  }
}

__global__ __launch_bounds__(NTHREADS) void attn_fused(
    const float* __restrict__ enc,   // [B,S,D]
    const float* __restrict__ dec,   // [B,D]
    float* __restrict__ out,         // [B,D] (used when direct==1)
    float* __restrict__ ws_acc,      // [B,C,D]
    float* __restrict__ ws_ml,       // [B,C,2]
    int C, int rowsPerChunk, int direct)
{
  extern __shared__ float smem[];
  float* e0 = smem;                  // TILE_W
  float* e1 = smem + TILE_W;         // TILE_W
  float* hl = smem + 2 * TILE_W;     // D_ (decoder hidden)
  float* sp = hl + D_;               // 8 waves x 16 partial scores

  const int tid  = threadIdx.x;
  const int lane = tid & 31;
  const int wave = tid >> 5;
  const int b     = blockIdx.y;
  const int chunk = blockIdx.x;
  const int row0  = chunk * rowsPerChunk;
  const int T     = rowsPerChunk / TILE_R;

  // Preload h[b,:] into LDS (reused 4096x; visibility covered by in-loop barrier).
  const float* hg = dec + (size_t)b * D_;
  #pragma unroll
  for (int i = 0; i < D_ / NTHREADS; ++i)
    hl[i * NTHREADS + tid] = hg[i * NTHREADS + tid];

  const float* encb = enc + ((size_t)b * S_ + row0) * D_;
  const unsigned e0o = lds_off(e0), e1o = lds_off(e1);

  async_tile_load(encb, e0o, tid);   // prologue: tile 0 in flight

  float m = -INFINITY, lsum = 0.f;
  float acc[8];
  #pragma unroll
  for (int j = 0; j < 8; ++j) acc[j] = 0.f;
  const int d0   = tid * 8;          // this thread's 8 output features
  const int hi   = lane >> 4;        // half-wave selector (K phase)
  const int mrow = lane & 15;        // WMMA-A row owned by this lane

  for (int t = 0; t < T; ++t) {
    const float* ecur = (t & 1) ? e1 : e0;
    if (t + 1 < T) {
      async_tile_load(encb + (size_t)(t + 1) * TILE_R * D_,
                      (t & 1) ? e0o : e1o, tid);
      // 32 new in flight; wait for the previous 32 (in-order completion).
      asm volatile("s_wait_asynccnt 32" ::: "memory");
    } else {
      asm volatile("s_wait_asynccnt 0" ::: "memory");
    }
    __syncthreads();                 // tile t visible to all waves

    // ---- score stage: 16-row dots via V_WMMA_F32_16X16X4_F32 ----
    // A[m,k] = E[row m, k] (ISA 16x4 f32 layout), B[k,n] = h[k] broadcast.
    // Wave w reduces K slice [w*256, w*256+256); D[m,*] = partial score of row m.
    v8f c = {};
    const float* arow = ecur + mrow * ROW_W + 2 * hi;
    const float* brow = hl + 2 * hi;
    const int kbase = wave * (D_ / 8);
    #pragma unroll 4
    for (int kk = 0; kk < D_ / 8; kk += 4) {
      int k = kbase + kk;
      v2f a  = *(const v2f*)(arow + k);
      v2f bb = *(const v2f*)(brow + k);
      c = __builtin_amdgcn_wmma_f32_16x16x4_f32(
              false, a, false, bb, (short)0, c, false, false);
    }
    // Lanes 0 and 16 hold rows 0-7 / 8-15 (N=0 column) -> stash partials.
    if (mrow == 0) {
      #pragma unroll
      for (int j = 0; j < 8; ++j) sp[wave * 16 + hi * 8 + j] = c[j];
    }
    __syncthreads();

    // ---- online softmax update (redundant per-thread, LDS broadcast reads) ----
    float s[16];
    #pragma unroll
    for (int r = 0; r < 16; ++r) {
      float a0 = sp[r];
      #pragma unroll
      for (int w = 1; w < 8; ++w) a0 += sp[w * 16 + r];
      s[r] = a0;
    }
    float mt = s[0];
    #pragma unroll
    for (int r = 1; r < 16; ++r) mt = fmaxf(mt, s[r]);
    float mnew  = fmaxf(m, mt);
    float scale = __expf(m - mnew);  // first tile: exp(-inf)=0
    float p[16], psum = 0.f;
    #pragma unroll
    for (int r = 0; r < 16; ++r) { p[r] = __expf(s[r] - mnew); psum += p[r]; }
    lsum = lsum * scale + psum;
    m = mnew;
    #pragma unroll
    for (int j = 0; j < 8; ++j) acc[j] *= scale;

    // ---- weighted accumulate: acc[d] += sum_r p[r] * E[r,d] ----
    const float* erow = ecur + d0;
    #pragma unroll
    for (int r = 0; r < 16; ++r) {
      v4f x0 = *(const v4f*)(erow + r * ROW_W);
      v4f x1 = *(const v4f*)(erow + r * ROW_W + 4);
      float pr = p[r];
      acc[0] = fmaf(pr, x0.x, acc[0]); acc[1] = fmaf(pr, x0.y, acc[1]);
      acc[2] = fmaf(pr, x0.z, acc[2]); acc[3] = fmaf(pr, x0.w, acc[3]);
      acc[4] = fmaf(pr, x1.x, acc[4]); acc[5] = fmaf(pr, x1.y, acc[5]);
      acc[6] = fmaf(pr, x1.z, acc[6]); acc[7] = fmaf(pr, x1.w, acc[7]);
    }
    __syncthreads();                 // done reading tile t; buffer reusable
  }

  if (direct) {
    float inv = 1.f / lsum;
    float* o = out + (size_t)b * D_ + d0;
    #pragma unroll
    for (int j = 0; j < 8; ++j) o[j] = acc[j] * inv;
  } else {
    float* wa = ws_acc + ((size_t)(b * C + chunk)) * D_ + d0;
    #pragma unroll
    for (int j = 0; j < 8; ++j) wa[j] = acc[j];
    if (tid == 0) {
      ws_ml[(b * C + chunk) * 2 + 0] = m;
      ws_ml[(b * C + chunk) * 2 + 1] = lsum;
    }
  }
}

// Merge per-chunk partials: out = sum_c exp(m_c - M) acc_c / sum_c exp(m_c - M) l_c
__global__ __launch_bounds__(NTHREADS) void attn_reduce(
    const float* __restrict__ ws_acc, const float* __restrict__ ws_ml,
    float* __restrict__ out, int C)
{
  const int b = blockIdx.x, tid = threadIdx.x, d0 = tid * 8;
  float M = -INFINITY;
  for (int c = 0; c < C; ++c) M = fmaxf(M, ws_ml[(b * C + c) * 2]);
  float L = 0.f;
  for (int c = 0; c < C; ++c)
    L += __expf(ws_ml[(b * C + c) * 2] - M) * ws_ml[(b * C + c) * 2 + 1];
  float o[8] = {0.f, 0.f, 0.f, 0.f, 0.f, 0.f, 0.f, 0.f};
  for (int c = 0; c < C; ++c) {
    float wc = __expf(ws_ml[(b * C + c) * 2] - M);
    const float* a = ws_acc + ((size_t)(b * C + c)) * D_ + d0;
    #pragma unroll
    for (int j = 0; j < 8; ++j) o[j] = fmaf(wc, a[j], o[j]);
  }
  float inv = 1.f / L;
  float* op = out + (size_t)b * D_ + d0;
  #pragma unroll
  for (int j = 0; j < 8; ++j) op[j] = o[j] * inv;
}

extern "C" void kernel_launch(void* const* d_in, const int* in_sizes, int n_in,
                              void* d_out, int out_size, void* d_ws, size_t ws_size,
                              hipStream_t stream) {
  (void)in_sizes; (void)n_in; (void)out_size;
  const float* dec = (const float*)d_in[0];   // decoder_hidden [B,D]
  const float* enc = (const float*)d_in[1];   // encoder_states [B,S,D]
  float* out = (float*)d_out;

  // Pick chunks-per-batch from available workspace (deterministic in ws_size).
  int C = 1;
  for (int c = 32; c >= 2; c >>= 1) {
    if ((size_t)B_ * (size_t)c * (size_t)(D_ + 2) * 4u <= ws_size) { C = c; break; }
  }
  const int direct = (C == 1);
  float* ws_acc = (float*)d_ws;
  float* ws_ml  = ws_acc + (size_t)B_ * C * D_;
  const int rowsPerChunk = S_ / C;
  const size_t shmem = (size_t)(2 * TILE_W + D_ + 128) * sizeof(float); // ~271 KB

  dim3 grid(C, B_), blk(NTHREADS);
  hipLaunchKernelGGL(attn_fused, grid, blk, shmem, stream,
                     enc, dec, out, ws_acc, ws_ml, C, rowsPerChunk, direct);
  if (!direct)
    hipLaunchKernelGGL(attn_reduce, dim3(B_), blk, 0, stream,
                       ws_acc, ws_ml, out, C);
}